// GCN_50895362457963
// MI455X (gfx1250) — compile-verified
//
#include <hip/hip_runtime.h>
#include <hip/hip_bf16.h>
#include <math.h>

// ---- WMMA fragment types (wave32, gfx1250) ----
typedef float v2f __attribute__((ext_vector_type(2)));
typedef float v8f __attribute__((ext_vector_type(8)));

// Hardware approximate ops (v_rsq_f32 / v_rcp_f32) — matches jax.lax.rsqrt
// semantics and avoids the IEEE div/sqrt expansion ladder.
__device__ __forceinline__ float fast_rsqrt(float x) { return __builtin_amdgcn_rsqf(x); }
__device__ __forceinline__ float fast_rcp(float x)   { return __builtin_amdgcn_rcpf(x); }

// ---------------------------------------------------------------------------
// Workspace layout (floats), all offsets from d_ws:
//   cnt  [N]    : float degree counter (atomic +1.0 per in-edge; exact ints)
//   dis  [N]    : deg^{-1/2} with self-loop
//   h1   [N,4]  : X @ W1
//   agg1 [N,4]  : accumulator, init h1*deg_inv + b1, then edge scatter-adds
//   h2   [N,2]  : relu(agg1) @ W2
// agg2 lives directly in d_out [N,2].
// Total: 12*N floats = 48 MB for N = 1e6.
// ---------------------------------------------------------------------------

__global__ void gcn_zero(float* __restrict__ p, int n) {
    int i = blockIdx.x * blockDim.x + threadIdx.x;
    int stride = gridDim.x * blockDim.x;
    for (; i < n; i += stride) p[i] = 0.0f;
}

// Degree count: atomicAdd of 1.0f is exact (counts << 2^24) and order-independent.
__global__ void gcn_deg(const int* __restrict__ dst, float* __restrict__ cnt, int e) {
    int i = blockIdx.x * blockDim.x + threadIdx.x;
    if (i < e) atomicAdd(&cnt[dst[i]], 1.0f);
}

// Layer-1 node transform: one wave handles 16 nodes via V_WMMA_F32_16X16X4_F32.
//   A (16x4): lanes 0-15 -> {K0,K1} of row M=lane; lanes 16-31 -> {K2,K3=pad} of row M=lane-16
//   B (4x16): VGPR0 -> rows K0 (lanes 0-15) / K2 (lanes 16-31); VGPR1 -> K1 / K3
//   D (16x16): VGPR v -> row v (lanes 0-15) / row v+8 (lanes 16-31); col = lane&15
// Only D columns 0..3 are meaningful (W1 is 3x4, zero-padded).
__global__ void gcn_node1(const float* __restrict__ x,
                          const float* __restrict__ W1, const float* __restrict__ b1,
                          const float* __restrict__ cnt,
                          float* __restrict__ dis,
                          float* __restrict__ h1, float* __restrict__ agg1,
                          int n) {
    const int lane = threadIdx.x & 31;
    const int wave = threadIdx.x >> 5;
    const int base = (blockIdx.x * (blockDim.x >> 5) + wave) << 4;   // 16 nodes / wave
    const int g  = lane >> 4;     // half-wave group
    const int hl = lane & 15;

    // A fragment: node features, K padded 3 -> 4 with zeros.
    const int  arow = base + hl;
    const bool aval = arow < n;
    v2f a;
    if (g == 0) {
        a.x = aval ? x[arow * 3 + 0] : 0.0f;
        a.y = aval ? x[arow * 3 + 1] : 0.0f;
    } else {
        a.x = aval ? x[arow * 3 + 2] : 0.0f;
        a.y = 0.0f;                               // K=3 pad
    }

    // B fragment: W1 (3x4 row-major), rows K, cols N (cols 4..15 zero).
    v2f b;
    b.x = (hl < 4) ? W1[(g ? 2 : 0) * 4 + hl] : 0.0f;                 // K=0 / K=2
    b.y = (g == 0 && hl < 4) ? W1[1 * 4 + hl] : 0.0f;                 // K=1 / K=3 pad

    v8f c = {};
    c = __builtin_amdgcn_wmma_f32_16x16x4_f32(false, a, false, b,
                                              (short)0, c, false, false);

    // Per-node deg^{-1/2} for this wave's 16 nodes (lanes 0-15): v_rsq_f32.
    if (g == 0 && aval) {
        dis[arow] = fast_rsqrt(1.0f + cnt[arow]);
    }

    // Scatter D: lane's column = hl; rows base + v + 8*g. Fold self-loop + bias
    // into the accumulator so the edge kernel only has to atomic-add.
    if (hl < 4) {
        const float bias = b1[hl];
        #pragma unroll
        for (int v = 0; v < 8; ++v) {
            int row = base + v + 8 * g;
            if (row < n) {
                float hv   = c[v];
                float dinv = fast_rcp(1.0f + cnt[row]);   // v_rcp_f32
                h1[row * 4 + hl]   = hv;
                agg1[row * 4 + hl] = hv * dinv + bias;
            }
        }
    }
}

// Layer-1 edge pass: gather h1[src] (b128), scale by dis[src]*dis[dst],
// scatter-add into agg1[dst]. All node arrays are L2-resident (<=16 MB each).
__global__ void gcn_edge1(const int* __restrict__ src, const int* __restrict__ dst,
                          const float* __restrict__ dis, const float* __restrict__ h1,
                          float* __restrict__ agg1, int e) {
    int i = blockIdx.x * blockDim.x + threadIdx.x;
    if (i >= e) return;
    int s = src[i], d = dst[i];
    float nrm = dis[s] * dis[d];
    const float4 h = *reinterpret_cast<const float4*>(h1 + 4 * s);
    float* o = agg1 + 4 * d;
    atomicAdd(o + 0, h.x * nrm);
    atomicAdd(o + 1, h.y * nrm);
    atomicAdd(o + 2, h.z * nrm);
    atomicAdd(o + 3, h.w * nrm);
}

// Layer-2 node transform: relu + (16x4)@(4x2) via the same WMMA (K=4 exact).
__global__ void gcn_node2(const float* __restrict__ agg1,
                          const float* __restrict__ W2, const float* __restrict__ b2,
                          const float* __restrict__ cnt,
                          float* __restrict__ h2, float* __restrict__ agg2,
                          int n) {
    const int lane = threadIdx.x & 31;
    const int wave = threadIdx.x >> 5;
    const int base = (blockIdx.x * (blockDim.x >> 5) + wave) << 4;
    const int g  = lane >> 4;
    const int hl = lane & 15;

    const int  arow = base + hl;
    const bool aval = arow < n;
    const int  k0   = g ? 2 : 0;
    v2f a;
    a.x = aval ? fmaxf(agg1[arow * 4 + k0 + 0], 0.0f) : 0.0f;  // relu fused
    a.y = aval ? fmaxf(agg1[arow * 4 + k0 + 1], 0.0f) : 0.0f;

    v2f b;                                                      // W2 is 4x2
    b.x = (hl < 2) ? W2[(k0 + 0) * 2 + hl] : 0.0f;
    b.y = (hl < 2) ? W2[(k0 + 1) * 2 + hl] : 0.0f;

    v8f c = {};
    c = __builtin_amdgcn_wmma_f32_16x16x4_f32(false, a, false, b,
                                              (short)0, c, false, false);

    if (hl < 2) {
        const float bias = b2[hl];
        #pragma unroll
        for (int v = 0; v < 8; ++v) {
            int row = base + v + 8 * g;
            if (row < n) {
                float hv   = c[v];
                float dinv = fast_rcp(1.0f + cnt[row]);       // v_rcp_f32
                h2[row * 2 + hl]   = hv;
                agg2[row * 2 + hl] = hv * dinv + bias;
            }
        }
    }
}

__global__ void gcn_edge2(const int* __restrict__ src, const int* __restrict__ dst,
                          const float* __restrict__ dis, const float* __restrict__ h2,
                          float* __restrict__ agg2, int e) {
    int i = blockIdx.x * blockDim.x + threadIdx.x;
    if (i >= e) return;
    int s = src[i], d = dst[i];
    float nrm = dis[s] * dis[d];
    const float2 h = *reinterpret_cast<const float2*>(h2 + 2 * s);
    float* o = agg2 + 2 * d;
    atomicAdd(o + 0, h.x * nrm);
    atomicAdd(o + 1, h.y * nrm);
}

// log_softmax over the 2 classes, in place on d_out.
__global__ void gcn_lsm(float* __restrict__ out, int n) {
    int i = blockIdx.x * blockDim.x + threadIdx.x;
    if (i >= n) return;
    float a0 = out[2 * i], a1 = out[2 * i + 1];
    float m  = fmaxf(a0, a1);
    float l  = m + logf(expf(a0 - m) + expf(a1 - m));
    out[2 * i]     = a0 - l;
    out[2 * i + 1] = a1 - l;
}

extern "C" void kernel_launch(void* const* d_in, const int* in_sizes, int n_in,
                              void* d_out, int out_size, void* d_ws, size_t ws_size,
                              hipStream_t stream) {
    const int n = in_sizes[0] / 3;       // x is [N,3]
    const int e = in_sizes[1] / 2;       // edge_index is [2,E]

    const float* x   = (const float*)d_in[0];
    const int*   ei  = (const int*)d_in[1];   // harness delivers integer inputs as int32
    const float* W1  = (const float*)d_in[2];
    const float* b1  = (const float*)d_in[3];
    const float* W2  = (const float*)d_in[4];
    const float* b2  = (const float*)d_in[5];
    const int* src = ei;
    const int* dst = ei + e;

    float* ws   = (float*)d_ws;
    float* cnt  = ws;                 // [N]
    float* dis  = ws + (size_t)n;     // [N]
    float* h1   = ws + (size_t)2 * n; // [N,4]
    float* agg1 = ws + (size_t)6 * n; // [N,4]
    float* h2   = ws + (size_t)10 * n;// [N,2]
    float* out  = (float*)d_out;      // [N,2] doubles as agg2

    const int TB = 256;
    const int egrid = (e + TB - 1) / TB;
    const int ngrid16 = (n + 127) / 128;      // 8 waves/block * 16 nodes/wave
    const int ngrid   = (n + TB - 1) / TB;

    gcn_zero <<<ngrid, TB, 0, stream>>>(cnt, n);
    gcn_deg  <<<egrid, TB, 0, stream>>>(dst, cnt, e);
    gcn_node1<<<ngrid16, TB, 0, stream>>>(x, W1, b1, cnt, dis, h1, agg1, n);
    gcn_edge1<<<egrid, TB, 0, stream>>>(src, dst, dis, h1, agg1, e);
    gcn_node2<<<ngrid16, TB, 0, stream>>>(agg1, W2, b2, cnt, h2, out, n);
    gcn_edge2<<<egrid, TB, 0, stream>>>(src, dst, dis, h2, out, e);
    gcn_lsm  <<<ngrid, TB, 0, stream>>>(out, n);
}